// Sparsemax_17497696764646
// MI455X (gfx1250) — compile-verified
//
#include <hip/hip_runtime.h>

#define D       4096
#define BLOCK   256
#define NWAVES  (BLOCK / 32)
#define CHUNKS  4          // 4 x float4 = 16 elements per thread

// Pointee type the async-to-LDS builtin expects (per hipcc diagnostic:
// "int __vector(4) *" with device/global and shared address spaces).
typedef int v4i __attribute__((vector_size(16)));
typedef __attribute__((address_space(1))) v4i gv4i;   // global (AS1)
typedef __attribute__((address_space(3))) v4i lv4i;   // LDS (AS3)

// ---------- wave32 reductions ----------
__device__ __forceinline__ float wave_sum_f(float v) {
#pragma unroll
  for (int m = 16; m >= 1; m >>= 1) v += __shfl_xor(v, m, 32);
  return v;
}
__device__ __forceinline__ int wave_sum_i(int v) {
#pragma unroll
  for (int m = 16; m >= 1; m >>= 1) v += __shfl_xor(v, m, 32);
  return v;
}

__global__ __launch_bounds__(BLOCK) void sparsemax_kernel(
    const float* __restrict__ x, float* __restrict__ out, int rows) {
  __shared__ float rowbuf[D];          // 16 KB row staging (async target)
  __shared__ float s_sum[NWAVES];
  __shared__ int   s_cnt[NWAVES];
  __shared__ float s_tau;
  __shared__ int   s_C;

  const int row = blockIdx.x;
  if (row >= rows) return;
  const int t = threadIdx.x;
  const float* rp = x + (size_t)row * D;

  float v[CHUNKS * 4];

#if __has_builtin(__builtin_amdgcn_global_load_async_to_lds_b128)
  // gfx1250 async path: stage this lane's 4x16B slices into LDS.
  // Each lane later re-reads exactly the bytes it issued, so a same-wave
  // s_wait_asynccnt(0) is the only ordering required (no barrier).
#pragma unroll
  for (int j = 0; j < CHUNKS; ++j) {
    const float* g = rp + j * 1024 + t * 4;
    float*       l = rowbuf + j * 1024 + t * 4;
    __builtin_amdgcn_global_load_async_to_lds_b128(
        (gv4i*)g, (lv4i*)l, 0, 0);
  }
#if __has_builtin(__builtin_amdgcn_s_wait_asynccnt)
  __builtin_amdgcn_s_wait_asynccnt(0);
#else
  asm volatile("s_wait_asynccnt 0" ::: "memory");
#endif
#pragma unroll
  for (int j = 0; j < CHUNKS; ++j) {
    float4 f = *reinterpret_cast<const float4*>(rowbuf + j * 1024 + t * 4);
    v[4 * j + 0] = f.x; v[4 * j + 1] = f.y;
    v[4 * j + 2] = f.z; v[4 * j + 3] = f.w;
  }
#else
  // Fallback: direct coalesced b128 global loads into registers.
#pragma unroll
  for (int j = 0; j < CHUNKS; ++j) {
    float4 f = *reinterpret_cast<const float4*>(rp + j * 1024 + t * 4);
    v[4 * j + 0] = f.x; v[4 * j + 1] = f.y;
    v[4 * j + 2] = f.z; v[4 * j + 3] = f.w;
  }
#endif

  // ---------- Michelot fixed-point iteration for tau ----------
  // tau = -inf  => first pass uses the full support: tau1 = (sum - 1)/D.
  float tau = -__builtin_inff();
#pragma unroll 1
  for (int it = 0; it < 64; ++it) {
    float s = 0.0f;
    int   c = 0;
#pragma unroll
    for (int k = 0; k < CHUNKS * 4; ++k) {
      if (v[k] > tau) { s += v[k]; ++c; }
    }
    s = wave_sum_f(s);
    c = wave_sum_i(c);
    const int wid = t >> 5, lane = t & 31;
    if (lane == 0) { s_sum[wid] = s; s_cnt[wid] = c; }
    __syncthreads();
    if (t == 0) {
      float S = 0.0f; int C = 0;
#pragma unroll
      for (int w = 0; w < NWAVES; ++w) { S += s_sum[w]; C += s_cnt[w]; }
      s_tau = (S - 1.0f) / (float)C;   // support always non-empty (rho >= 1)
      s_C   = C;
    }
    __syncthreads();
    const float tau_new = s_tau;
    // Fixed point: same active set -> bitwise-identical tau (deterministic
    // reduction order). Uniform across the block.
    if (tau_new == tau) break;
    tau = tau_new;
    __syncthreads();   // protect s_sum/s_cnt reuse next iteration
  }

  // ---------- project & store (coalesced b128) ----------
  float* op = out + (size_t)row * D;
#pragma unroll
  for (int j = 0; j < CHUNKS; ++j) {
    float4 o;
    o.x = fmaxf(v[4 * j + 0] - tau, 0.0f);
    o.y = fmaxf(v[4 * j + 1] - tau, 0.0f);
    o.z = fmaxf(v[4 * j + 2] - tau, 0.0f);
    o.w = fmaxf(v[4 * j + 3] - tau, 0.0f);
    *reinterpret_cast<float4*>(op + j * 1024 + t * 4) = o;
  }
}

extern "C" void kernel_launch(void* const* d_in, const int* in_sizes, int n_in,
                              void* d_out, int out_size, void* d_ws, size_t ws_size,
                              hipStream_t stream) {
  const float* x = (const float*)d_in[0];
  float* out = (float*)d_out;
  const int rows = in_sizes[0] / D;   // 8192
  sparsemax_kernel<<<dim3(rows), dim3(BLOCK), 0, stream>>>(x, out, rows);
}